// MLMMasker_6347961663777
// MI455X (gfx1250) — compile-verified
//
#include <hip/hip_runtime.h>
#include <stdint.h>

typedef int      v4i __attribute__((ext_vector_type(4)));
typedef uint32_t u32;
typedef uint64_t u64;

#define SP_REG 8   // special tokens held in SGPRs (wave-uniform); cold loop beyond

// ---------------------------------------------------------------------------
// Philox4x32-7: counter-based RNG, deterministic in the counter => identical
// across graph replays. One call yields 128 bits = entropy for TWO tokens:
//   token k uses words (2k): [31:16] inclusion draw, [15:0] mask draw
//              and   (2k+1): bit 0 coin, top bits -> rand index (umulhi)
// 16-bit threshold resolution quantizes p by < 1.6e-5 (truncated) (≪ sampling noise of the
// 1M-draw Bernoulli stream); coin/index bit ranges are disjoint.
// ---------------------------------------------------------------------------
__device__ __forceinline__ void philox4x32_7(u32 ctr, u32& o0, u32& o1, u32& o2, u32& o3) {
    u32 x0 = ctr, x1 = 0x55AA33CCu, x2 = 0x2C3E4F5Au, x3 = 0x6A09E667u;
    u32 k0 = 42u, k1 = 0xBB67AE85u;
#pragma unroll
    for (int r = 0; r < 7; ++r) {
        u32 hi0 = __umulhi(0xD2511F53u, x0);
        u32 lo0 = 0xD2511F53u * x0;
        u32 hi1 = __umulhi(0xCD9E8D57u, x2);
        u32 lo1 = 0xCD9E8D57u * x2;
        x0 = hi1 ^ x1 ^ k0;  x1 = lo1;
        x2 = hi0 ^ x3 ^ k1;  x3 = lo0;
        k0 += 0x9E3779B9u;   k1 += 0xBB67AE85u;
    }
    o0 = x0; o1 = x1; o2 = x2; o3 = x3;
}

// Per-token decision from two random words (shared by vector body and tail).
__device__ __forceinline__ void mlm_decide(u32 wa, u32 wb, int id, int lb,
                                           const int (&sp)[SP_REG], int n_special,
                                           bool has_special,
                                           const int* __restrict__ special_tokens,
                                           const int* __restrict__ standard_tokens,
                                           u32 n_standard, bool has_standard,
                                           int mask_tok, u32 thr_incl, u32 thr_mask,
                                           int& out_id, int& out_lb) {
    // Special check vs SGPR-resident tokens: v_cmp against scalar operands,
    // lane-mask combines fold to s_or_b32. No LDS traffic.
    bool special = false;
#pragma unroll
    for (int s = 0; s < SP_REG; ++s) special |= (id == sp[s]);
    special = special && has_special;
    for (int s = SP_REG; s < n_special; ++s)       // cold generic path (ns > 8)
        special |= (id == special_tokens[s]);

    const bool incl  = (!special) && ((wa >> 16)     < thr_incl);
    const bool rmask = incl && ((wa & 0xFFFFu) < thr_mask);
    const bool rrand = incl && !rmask && (wb & 1u) && has_standard;

    int tok = id;
    if (rrand) {
        // floor(u * n) via 32x32->64 mulhi (top bits; disjoint from coin bit).
        // Exec-masked gather, ~1.5% of lanes; 131 KB table is L2-resident.
        u32 ridx = (u32)(((u64)wb * (u64)n_standard) >> 32);
        tok = standard_tokens[ridx];
    }
    out_id = rmask ? mask_tok : tok;
    out_lb = incl ? lb : -100;
}

__global__ __launch_bounds__(256) void mlm_mask_kernel(
    const int* __restrict__ input_ids,
    const int* __restrict__ labels,
    const float* __restrict__ mask_prob_p,
    const float* __restrict__ keep_replace_p,
    const int* __restrict__ special_tokens,
    const int* __restrict__ standard_tokens,
    const int* __restrict__ mask_token_p,
    int* __restrict__ out_ids,
    int* __restrict__ out_labels,
    int n, int n_special, int n_standard)
{
    // --- Uniform setup (scalar, once per wave) ------------------------------
    const float mp        = mask_prob_p[0];
    const float krp       = keep_replace_p[0];
    const float mlm_prob  = mp + 2.0f * krp;
    const float mask_port = mp / mlm_prob;
    const int   mask_tok  = mask_token_p[0];

    // 16-bit integer thresholds; value 65536 makes p>=1 always-true since the
    // tested halfword is <= 65535. Double math avoids float rounding corners.
    const u32 thr_incl = (u32)fmin((double)mlm_prob  * 65536.0, 65536.0);
    const u32 thr_mask = (u32)fmin((double)mask_port * 65536.0, 65536.0);

    const bool has_special  = (n_special  > 0);
    const bool has_standard = (n_standard > 0);

    // Special tokens -> SGPRs. Single branch region so the 8 uniform loads
    // can issue as one s_load clause (index clamped, padding = last token).
    int sp[SP_REG];
    if (has_special) {
        const int last = n_special - 1;
#pragma unroll
        for (int s = 0; s < SP_REG; ++s) {
            int idx = (s < n_special) ? s : last;
            sp[s] = special_tokens[idx];
        }
    } else {
#pragma unroll
        for (int s = 0; s < SP_REG; ++s) sp[s] = 0;   // gated by has_special
    }

    const int nvec   = n >> 2;                       // int4 granularity
    const int stride = gridDim.x * blockDim.x;

    for (int v = blockIdx.x * blockDim.x + threadIdx.x; v < nvec; v += stride) {
        // Prefetch next grid-stride tile (speculative, dropped if OOB)
        // -> global_prefetch_b8 on gfx1250.
        __builtin_prefetch(input_ids + (u64)(v + stride) * 4, 0, 1);
        __builtin_prefetch(labels    + (u64)(v + stride) * 4, 0, 1);

        // Streaming 128-bit NT loads: read-once data, don't pollute caches.
        v4i ids = __builtin_nontemporal_load((const v4i*)input_ids + v);
        v4i lbs = __builtin_nontemporal_load((const v4i*)labels    + v);

        // Two Philox calls cover all four tokens (2 words/token).
        u32 w[8];
        philox4x32_7((u32)(v << 1) | 0u, w[0], w[1], w[2], w[3]);
        philox4x32_7((u32)(v << 1) | 1u, w[4], w[5], w[6], w[7]);

        v4i oid, olb;
#pragma unroll
        for (int j = 0; j < 4; ++j) {
            int oi, ol;
            mlm_decide(w[2 * j], w[2 * j + 1], ids[j], lbs[j],
                       sp, n_special, has_special,
                       special_tokens, standard_tokens,
                       (u32)n_standard, has_standard,
                       mask_tok, thr_incl, thr_mask, oi, ol);
            oid[j] = oi;
            olb[j] = ol;
        }

        // Streaming 128-bit NT stores.
        __builtin_nontemporal_store(oid, (v4i*)out_ids    + v);
        __builtin_nontemporal_store(olb, (v4i*)out_labels + v);
    }

    // Scalar tail (N % 4 != 0); counter space disjoint from the vector path.
    if (blockIdx.x == 0) {
        int tail = n & 3;
        if ((int)threadIdx.x < tail) {
            int i = (nvec << 2) + (int)threadIdx.x;
            u32 wa, wb, t2, t3;
            philox4x32_7(0x80000000u + (u32)i, wa, wb, t2, t3);
            int oi, ol;
            mlm_decide(wa, wb, input_ids[i], labels[i],
                       sp, n_special, has_special,
                       special_tokens, standard_tokens,
                       (u32)n_standard, has_standard,
                       mask_tok, thr_incl, thr_mask, oi, ol);
            out_ids[i]    = oi;
            out_labels[i] = ol;
        }
    }
}

extern "C" void kernel_launch(void* const* d_in, const int* in_sizes, int n_in,
                              void* d_out, int out_size, void* d_ws, size_t ws_size,
                              hipStream_t stream) {
    (void)n_in; (void)d_ws; (void)ws_size; (void)out_size;

    const int*   input_ids       = (const int*)  d_in[0];
    const int*   labels          = (const int*)  d_in[1];
    const float* mask_prob       = (const float*)d_in[2];
    const float* keep_replace    = (const float*)d_in[3];
    const int*   special_tokens  = (const int*)  d_in[4];
    const int*   standard_tokens = (const int*)  d_in[5];
    const int*   mask_token_id   = (const int*)  d_in[6];

    const int n          = in_sizes[0];      // B*S tokens
    const int n_special  = in_sizes[4];
    const int n_standard = in_sizes[5];

    int* out_ids    = (int*)d_out;           // outputs concatenated flat
    int* out_labels = out_ids + n;           // in reference return order

    const int threads = 256;                 // 8 wave32 waves per block
    int nvec  = (n + 3) >> 2;
    int grid  = (nvec + threads - 1) / threads;
    if (grid > 4096) grid = 4096;            // grid-stride handles the rest
    if (grid < 1)    grid = 1;

    mlm_mask_kernel<<<grid, threads, 0, stream>>>(
        input_ids, labels, mask_prob, keep_replace,
        special_tokens, standard_tokens, mask_token_id,
        out_ids, out_labels, n, n_special, n_standard);
}